// GCN_59279138619891
// MI455X (gfx1250) — compile-verified
//
#include <hip/hip_runtime.h>
#include <hip/hip_bf16.h>

#define N_NODES 100000
#define N_EDGES 3200000
#define N_FEAT  512
#define HID     16
#define XS_PITCH 68
#define CHUNK   64
#define KCHUNKS 8
#define PBLOCKS 512

typedef __attribute__((ext_vector_type(2))) float v2f;
typedef __attribute__((ext_vector_type(4))) float v4f;
typedef __attribute__((ext_vector_type(8))) float v8f;

// async global -> LDS copy of 16 bytes (gfx1250, ASYNCcnt-tracked)
__device__ __forceinline__ void async_copy16(unsigned lds_addr, const void* gaddr) {
    asm volatile("global_load_async_to_lds_b128 %0, %1, off"
                 :: "v"(lds_addr), "v"((unsigned long long)gaddr)
                 : "memory");
}

// ---------------- init: deg=1 (self loop), zero atomic accumulators ----------------
__global__ void k_init(float* __restrict__ deg, float* __restrict__ a1, float* __restrict__ a2) {
    int gid = blockIdx.x * blockDim.x + threadIdx.x;
    if (gid < N_NODES * HID) { a1[gid] = 0.0f; a2[gid] = 0.0f; }
    if (gid < N_NODES) deg[gid] = 1.0f;
}

// ---------------- in-degree over dst ----------------
__global__ void k_deg(const int* __restrict__ dst, float* __restrict__ deg) {
    int e = blockIdx.x * blockDim.x + threadIdx.x;
    if (e < N_EDGES) atomicAdd(&deg[dst[e]], 1.0f);
}

// ---------------- dinv = rsqrt(deg), in place ----------------
__global__ void k_dinv(float* __restrict__ deg) {
    int i = blockIdx.x * blockDim.x + threadIdx.x;
    if (i < N_NODES) {
        float d = deg[i];
        deg[i] = d > 0.0f ? rsqrtf(d) : 0.0f;
    }
}

// ---------------- h1 = x @ W1, WMMA f32 16x16x4, double-buffered async LDS staging ----
__global__ __launch_bounds__(256) void k_gemm1(const float* __restrict__ x,
                                               const float* __restrict__ W1,
                                               float* __restrict__ h1) {
    __shared__ float xs[2][128 * XS_PITCH];   // two 128x64 chunks of x (padded pitch)
    __shared__ float wsm[2][CHUNK * 16];      // two 64x16 chunks of W1
    const int t    = threadIdx.x;
    const int wave = t >> 5;
    const int lane = t & 31;
    const int half = lane >> 4;               // 0: K pair {0,1}; 1: K pair {2,3}
    const int ml   = lane & 15;
    const int row0 = blockIdx.x * 128;

    // issue one chunk's async copies: 9 ASYNCcnt ops per wave (8 x-tiles + 1 W-tile)
    auto issue = [&](int kc, int buf) {
        #pragma unroll
        for (int j = 0; j < 8; ++j) {
            int idx = j * 256 + t;
            int r   = idx >> 4;                 // 0..127
            int c4  = (idx & 15) << 2;          // 0..60 step 4
            int row = row0 + r;
            if (row >= N_NODES) row = N_NODES - 1;   // clamp: tail rows masked at store
            unsigned lds = (unsigned)(size_t)(&xs[buf][r * XS_PITCH + c4]);
            async_copy16(lds, x + (size_t)row * N_FEAT + kc * CHUNK + c4);
        }
        int kr = t >> 2;
        int c4 = (t & 3) << 2;
        unsigned lds = (unsigned)(size_t)(&wsm[buf][kr * 16 + c4]);
        async_copy16(lds, W1 + (size_t)(kc * CHUNK + kr) * 16 + c4);
    };

    issue(0, 0);
    v8f acc = {};
    for (int kc = 0; kc < KCHUNKS; ++kc) {
        const int buf = kc & 1;
        if (kc + 1 < KCHUNKS) {
            issue(kc + 1, buf ^ 1);                    // prefetch next chunk
            asm volatile("s_wait_asynccnt 0x9" ::: "memory");  // retire current chunk only
        } else {
            asm volatile("s_wait_asynccnt 0x0" ::: "memory");
        }
        __syncthreads();

        const float* xrow = &xs[buf][(wave * 16 + ml) * XS_PITCH];
        const float* wb   = &wsm[buf][0];
        #pragma unroll
        for (int kk = 0; kk < CHUNK / 4; ++kk) {
            int kb = kk * 4 + 2 * half;
            v2f a = { xrow[kb], xrow[kb + 1] };
            v2f b = { wb[kb * 16 + ml], wb[(kb + 1) * 16 + ml] };
            acc = __builtin_amdgcn_wmma_f32_16x16x4_f32(
                false, a, false, b, (short)0, acc, false, false);
        }
        __syncthreads();   // buffer may be overwritten by next iteration's issue
    }
    // C/D layout: lanes 0-15: N=lane, M=vgpr; lanes 16-31: N=lane-16, M=8+vgpr
    #pragma unroll
    for (int i = 0; i < 8; ++i) {
        int m   = half * 8 + i;
        int row = row0 + wave * 16 + m;
        if (row < N_NODES) h1[(size_t)row * HID + ml] = acc[i];
    }
}

// ---------------- edge scatter: a[dst] += dinv[s]*dinv[d] * h[src] ----------------
__global__ void k_scatter(const int* __restrict__ src, const int* __restrict__ dst,
                          const float* __restrict__ dinv, const float* __restrict__ h,
                          float* __restrict__ a) {
    long long gid = (long long)blockIdx.x * blockDim.x + threadIdx.x;
    if (gid >= (long long)N_EDGES * HID) return;
    int e = (int)(gid >> 4);
    int f = (int)(gid & 15);
    int s = src[e], d = dst[e];
    float norm = dinv[s] * dinv[d];
    atomicAdd(&a[(size_t)d * HID + f], norm * h[(size_t)s * HID + f]);
}

// ---------------- self loop + bias + relu; y1 written over h1 ----------------
__global__ void k_post1(float* __restrict__ h1, const float* __restrict__ a1,
                        const float* __restrict__ dinv, const float* __restrict__ b1) {
    int gid = blockIdx.x * blockDim.x + threadIdx.x;
    if (gid >= N_NODES * HID) return;
    int i = gid >> 4, f = gid & 15;
    float di = dinv[i];
    float v  = a1[gid] + di * di * h1[gid] + b1[f];
    h1[gid]  = v > 0.0f ? v : 0.0f;
}

// ---------------- h2 = y1 @ W2 (K=16): 4 WMMAs per wave, straight from L2 ----------------
__global__ __launch_bounds__(256) void k_gemm2(const float* __restrict__ y1,
                                               const float* __restrict__ W2,
                                               float* __restrict__ h2) {
    const int t    = threadIdx.x;
    const int wave = t >> 5;
    const int lane = t & 31;
    const int half = lane >> 4;
    const int ml   = lane & 15;
    const int rowBase = blockIdx.x * 128 + wave * 16;
    int rowA = rowBase + ml;
    if (rowA >= N_NODES) rowA = N_NODES - 1;   // clamp: keep EXEC full

    v8f acc = {};
    #pragma unroll
    for (int kk = 0; kk < 4; ++kk) {
        int kb = kk * 4 + 2 * half;
        v2f a = { y1[(size_t)rowA * HID + kb], y1[(size_t)rowA * HID + kb + 1] };
        v2f b = { W2[kb * 16 + ml], W2[(kb + 1) * 16 + ml] };
        acc = __builtin_amdgcn_wmma_f32_16x16x4_f32(
            false, a, false, b, (short)0, acc, false, false);
    }
    #pragma unroll
    for (int i = 0; i < 8; ++i) {
        int m   = half * 8 + i;
        int row = rowBase + m;
        if (row < N_NODES) h2[(size_t)row * HID + ml] = acc[i];
    }
}

// ---------------- pooled partial sums: per-block LDS reduction over nodes ----------------
__global__ __launch_bounds__(256) void k_pool(const float* __restrict__ h2,
                                              const float* __restrict__ a2,
                                              const float* __restrict__ dinv,
                                              const float* __restrict__ b2,
                                              float* __restrict__ partials) {
    __shared__ float red[256];
    int t = threadIdx.x;
    int f = t & 15, r0 = t >> 4;
    float bf  = b2[f];
    float acc = 0.0f;
    for (int i = blockIdx.x * 16 + r0; i < N_NODES; i += PBLOCKS * 16) {
        float di = dinv[i];
        acc += a2[(size_t)i * HID + f] + di * di * h2[(size_t)i * HID + f] + bf;
    }
    red[t] = acc;
    __syncthreads();
    for (int s = 128; s >= 16; s >>= 1) {    // strides multiples of 16 -> same f
        if (t < s) red[t] += red[t + s];
        __syncthreads();
    }
    if (t < 16) partials[blockIdx.x * 16 + t] = red[t];
}

// ---------------- final: sum partials, mean, softmax -> out[16] ----------------
__global__ __launch_bounds__(256) void k_final(const float* __restrict__ partials,
                                               float* __restrict__ out) {
    __shared__ float red[256];
    __shared__ float sm[16];
    int t = threadIdx.x;
    int f = t & 15, j0 = t >> 4;
    float acc = 0.0f;
    for (int j = j0; j < PBLOCKS; j += 16) acc += partials[j * 16 + f];
    red[t] = acc;
    __syncthreads();
    for (int s = 128; s >= 16; s >>= 1) {
        if (t < s) red[t] += red[t + s];
        __syncthreads();
    }
    if (t < 16) sm[t] = red[t] * (1.0f / (float)N_NODES);
    __syncthreads();
    if (t == 0) {
        float m = sm[0];
        for (int i = 1; i < 16; ++i) m = fmaxf(m, sm[i]);
        float e[16], ssum = 0.0f;
        for (int i = 0; i < 16; ++i) { e[i] = __expf(sm[i] - m); ssum += e[i]; }
        float inv = 1.0f / ssum;
        for (int i = 0; i < 16; ++i) out[i] = e[i] * inv;
    }
}

extern "C" void kernel_launch(void* const* d_in, const int* in_sizes, int n_in,
                              void* d_out, int out_size, void* d_ws, size_t ws_size,
                              hipStream_t stream) {
    const float* x  = (const float*)d_in[0];
    const int*   ei = (const int*)d_in[1];
    const float* W1 = (const float*)d_in[2];
    const float* b1 = (const float*)d_in[3];
    const float* W2 = (const float*)d_in[4];
    const float* b2 = (const float*)d_in[5];
    const int* src = ei;
    const int* dst = ei + N_EDGES;

    float* ws  = (float*)d_ws;
    float* deg = ws;                               // N floats, becomes dinv in place
    float* h1  = deg + N_NODES;                    // N*16 (gemm1 out, then relu'd y1)
    float* a1  = h1  + (size_t)N_NODES * HID;      // N*16 atomic accum layer 1
    float* h2  = a1  + (size_t)N_NODES * HID;      // N*16 gemm2 out
    float* a2  = h2  + (size_t)N_NODES * HID;      // N*16 atomic accum layer 2
    float* partials = a2 + (size_t)N_NODES * HID;  // PBLOCKS*16
    float* out = (float*)d_out;

    const int nf   = N_NODES * HID;
    const int gNF  = (nf + 255) / 256;
    const int gN   = (N_NODES + 255) / 256;
    const int gE   = (N_EDGES + 255) / 256;
    const int gEF  = (int)(((long long)N_EDGES * HID + 255) / 256);
    const int gRow = (N_NODES + 127) / 128;

    k_init   <<<gNF, 256, 0, stream>>>(deg, a1, a2);
    k_deg    <<<gE,  256, 0, stream>>>(dst, deg);
    k_dinv   <<<gN,  256, 0, stream>>>(deg);
    k_gemm1  <<<gRow,256, 0, stream>>>(x, W1, h1);
    k_scatter<<<gEF, 256, 0, stream>>>(src, dst, deg, h1, a1);
    k_post1  <<<gNF, 256, 0, stream>>>(h1, a1, deg, b1);
    k_gemm2  <<<gRow,256, 0, stream>>>(h1, W2, h2);
    k_scatter<<<gEF, 256, 0, stream>>>(src, dst, deg, h2, a2);
    k_pool   <<<PBLOCKS, 256, 0, stream>>>(h2, a2, deg, b2, partials);
    k_final  <<<1,   256, 0, stream>>>(partials, out);
}